// FastWeightsClassifier_82540681494708
// MI455X (gfx1250) — compile-verified
//
#include <hip/hip_runtime.h>

#define R_DIM 256
#define EMB_DIM 128
#define B_DIM 64
#define T_DIM 128
#define HEAD_DIM 100
#define NCLS 10
#define ETA 0.5f
#define LAM 0.9f
#define LN_EPS 1e-5f
#define APAD 260   // row pitch: 1040 B -> 16B-aligned rows, uniform LDS bank use

typedef __attribute__((ext_vector_type(2))) float v2f;
typedef __attribute__((ext_vector_type(8))) float v8f;

// ---------------------------------------------------------------------------
// Kernel 1: Bnd[t*64+b, r] = sum_k emb[x_ids[b,t], k] * C_w[r, k] + bias[r]
// 8192 x 256 x 128 f32 GEMM via V_WMMA_F32_16X16X4_F32, one 16x16 tile / wave.
// ---------------------------------------------------------------------------
__global__ __launch_bounds__(128) void boundary_gemm_wmma(
    const int* __restrict__ x_ids, const float* __restrict__ emb,
    const float* __restrict__ C_w, const float* __restrict__ bias,
    float* __restrict__ Bnd)
{
  const int lane = threadIdx.x & 31;
  const int wid  = threadIdx.x >> 5;
  const int tile = blockIdx.x * 4 + wid;   // 8192 tiles = 512 (M) x 16 (N)
  const int mt = tile >> 4;
  const int nt = tile & 15;
  const int m0 = mt << 4;
  const int n0 = nt << 4;

  const int half = lane >> 4;              // 0: K=0,1   1: K=2,3
  const int l16  = lane & 15;
  const int koff = half << 1;

  // A operand (16x4 f32): lane = row M, 2 VGPRs = two K values
  const int m   = m0 + l16;
  const int t   = m >> 6;                  // m = t*64 + b
  const int bb  = m & 63;
  const int tok = x_ids[bb * T_DIM + t];
  const float* arow = emb + (long)tok * EMB_DIM + koff;
  // B operand (4x16 f32): B[k][n] = C_w[n][k]; lane = col N, 2 VGPRs = two K
  const float* brow = C_w + (long)(n0 + l16) * EMB_DIM + koff;

  v8f c = {0.f, 0.f, 0.f, 0.f, 0.f, 0.f, 0.f, 0.f};
#pragma unroll
  for (int k = 0; k < EMB_DIM; k += 4) {
    v2f a  = *reinterpret_cast<const v2f*>(arow + k);
    v2f bv = *reinterpret_cast<const v2f*>(brow + k);
    c = __builtin_amdgcn_wmma_f32_16x16x4_f32(false, a, false, bv,
                                              (short)0, c, false, false);
  }

  // C/D layout: VGPR i: lanes 0-15 -> M=i, lanes 16-31 -> M=8+i; N = lane%16
  const int nn = n0 + l16;
  const float bn = bias[nn];
#pragma unroll
  for (int i = 0; i < 8; ++i) {
    int mm = m0 + i + (half << 3);
    Bnd[(long)mm * R_DIM + nn] = c[i] + bn;
  }
}

// ---------------------------------------------------------------------------
// Block LayerNorm statistics over 256 values (1 per thread), 8 wave32s.
// ---------------------------------------------------------------------------
__device__ inline void blk_stats(float v, float* red, float& mean, float& rstd)
{
  float s = v, q = v * v;
#pragma unroll
  for (int o = 16; o > 0; o >>= 1) {
    s += __shfl_xor(s, o, 32);
    q += __shfl_xor(q, o, 32);
  }
  const int wid = threadIdx.x >> 5;
  __syncthreads();                              // protect red[] reuse
  if ((threadIdx.x & 31) == 0) { red[wid] = s; red[8 + wid] = q; }
  __syncthreads();
  float S = 0.f, Q = 0.f;
#pragma unroll
  for (int i = 0; i < 8; ++i) { S += red[i]; Q += red[8 + i]; }
  mean = S * (1.0f / R_DIM);
  float var = Q * (1.0f / R_DIM) - mean * mean;
  rstd = __frsqrt_rn(var + LN_EPS);
}

// ---------------------------------------------------------------------------
// Kernel 2: one workgroup per batch sample. A (256 x 260 padded, 260 KB)
// lives in LDS for all 128 timesteps; thread r owns row A[r,:]. All A-row and
// broadcast-vector traffic is float4 -> ds_load_b128 / ds_store_b128.
// ---------------------------------------------------------------------------
__global__ __launch_bounds__(256) void fastweights_recurrent(
    const float* __restrict__ Bnd, const float* __restrict__ W,
    const float* __restrict__ ln_w, const float* __restrict__ ln_b,
    const float* __restrict__ h1w,  const float* __restrict__ h1b,
    const float* __restrict__ h2w,  const float* __restrict__ h2b,
    float* __restrict__ out)
{
  extern __shared__ float smem[];
  float* Amat = smem;                           // 256 * 260 (16B-aligned rows)
  float* h_s  = Amat + R_DIM * APAD;            // 256
  float* hs_s = h_s + R_DIM;                    // 256
  float* hid  = hs_s + R_DIM;                   // 128
  float* red  = hid + 128;                      // 16

  const int r = threadIdx.x;
  const int bsample = blockIdx.x;

  float* arow = Amat + r * APAD;
  float4* a4 = reinterpret_cast<float4*>(arow);
  const float4* h4  = reinterpret_cast<const float4*>(h_s);
  const float4* hs4 = reinterpret_cast<const float4*>(hs_s);

#pragma unroll 8
  for (int s = 0; s < R_DIM / 4; ++s) a4[s] = make_float4(0.f, 0.f, 0.f, 0.f);
  h_s[r] = 0.f;
  const float lw = ln_w[r];
  const float lb = ln_b[r];
  const float* wrow = W + (long)r * R_DIM;
  __syncthreads();

  for (int t = 0; t < T_DIM; ++t) {
    // boundary[r] = precomputed (e@C^T + b) + (W h)[r]
    float bval = Bnd[(long)(t * B_DIM + bsample) * R_DIM + r];
    const float4* w4 = reinterpret_cast<const float4*>(wrow);
#pragma unroll 4
    for (int k = 0; k < R_DIM / 4; ++k) {
      float4 w = w4[k];
      float4 h = h4[k];
      bval += w.x * h.x + w.y * h.y + w.z * h.z + w.w * h.w;
    }
    // hs0 = relu(LN(boundary))
    float mu, rs;
    blk_stats(bval, red, mu, rs);
    float hs0 = fmaxf((bval - mu) * rs * lw + lb, 0.f);
    hs_s[r] = hs0;
    __syncthreads();
    // Ah[r] = sum_s A[r,s] * hs0[s]  (row r is thread-private, b128 DS ops)
    float ah = 0.f;
#pragma unroll 4
    for (int k = 0; k < R_DIM / 4; ++k) {
      float4 av = a4[k];
      float4 hv = hs4[k];
      ah += av.x * hv.x + av.y * hv.y + av.z * hv.z + av.w * hv.w;
    }
    float x2 = bval + ah;
    blk_stats(x2, red, mu, rs);      // its barriers also fence the Ah reads
    float hs1 = fmaxf((x2 - mu) * rs * lw + lb, 0.f);
    hs_s[r] = hs1;
    h_s[r]  = hs1;
    __syncthreads();
    // A = LAM*A + ETA * hs1 hs1^T   (row r private; hs broadcast b128 reads)
    const float er = ETA * hs1;
#pragma unroll 4
    for (int k = 0; k < R_DIM / 4; ++k) {
      float4 av = a4[k];
      float4 hv = hs4[k];
      av.x = LAM * av.x + er * hv.x;
      av.y = LAM * av.y + er * hv.y;
      av.z = LAM * av.z + er * hv.z;
      av.w = LAM * av.w + er * hv.w;
      a4[k] = av;
    }
    __syncthreads();
  }

  // Head: hidden = relu(h @ h1w^T + h1b); logits = hidden @ h2w^T + h2b
  if (r < HEAD_DIM) {
    float a = h1b[r];
    const float4* hw = reinterpret_cast<const float4*>(h1w + (long)r * R_DIM);
#pragma unroll 4
    for (int k = 0; k < R_DIM / 4; ++k) {
      float4 w = hw[k];
      float4 h = h4[k];
      a += w.x * h.x + w.y * h.y + w.z * h.z + w.w * h.w;
    }
    hid[r] = fmaxf(a, 0.f);
  }
  __syncthreads();
  if (r < NCLS) {
    float a = h2b[r];
    const float* hw = h2w + (long)r * HEAD_DIM;
#pragma unroll 4
    for (int j = 0; j < HEAD_DIM; ++j) a += hw[j] * hid[j];
    out[bsample * NCLS + r] = a;
  }
}

// ---------------------------------------------------------------------------
extern "C" void kernel_launch(void* const* d_in, const int* in_sizes, int n_in,
                              void* d_out, int out_size, void* d_ws, size_t ws_size,
                              hipStream_t stream)
{
  const int*   x_ids = (const int*)  d_in[0];
  const float* emb   = (const float*)d_in[1];
  const float* W     = (const float*)d_in[2];
  const float* C_w   = (const float*)d_in[3];
  const float* bvec  = (const float*)d_in[4];
  const float* ln_w  = (const float*)d_in[5];
  const float* ln_b  = (const float*)d_in[6];
  const float* h1w   = (const float*)d_in[7];
  const float* h1b   = (const float*)d_in[8];
  const float* h2w   = (const float*)d_in[9];
  const float* h2b   = (const float*)d_in[10];
  float* out = (float*)d_out;
  float* Bnd = (float*)d_ws;   // (T*B, R) f32 = 8 MB scratch

  // 8192 wave-tiles of 16x16, 4 waves / block
  boundary_gemm_wmma<<<2048, 128, 0, stream>>>(x_ids, emb, C_w, bvec, Bnd);

  size_t lds_bytes = (size_t)(R_DIM * APAD + 2 * R_DIM + 128 + 16) * sizeof(float);
  hipFuncSetAttribute(reinterpret_cast<const void*>(fastweights_recurrent),
                      hipFuncAttributeMaxDynamicSharedMemorySize, (int)lds_bytes);
  fastweights_recurrent<<<64, 256, lds_bytes, stream>>>(
      Bnd, W, ln_w, ln_b, h1w, h1b, h2w, h2b, out);
}